// Conv2DTransposeNN_78305843740719
// MI455X (gfx1250) — compile-verified
//
#include <hip/hip_runtime.h>

// Nearest-neighbor 2x2 upsample: x [4, 384, 384, 64] f32 -> out [4, 768, 768, 64] f32
//   out[b, p, q, c] = x[b, p>>1, q>>1, c]
//
// Pure data movement: 151 MB read + 604 MB written, 0 FLOPs -> HBM-bound,
// ~32 us floor at 23.3 TB/s. Kernel issues exactly the mandatory bytes:
// one B128 NT load + four B128 NT stores per thread (full 128B cachelines).
//
// Round-2 refinement: 2D grid (blockIdx.y = input row index b*H+i) removes
// all integer div/mod from the address math. Key identity: the output row
// index is b*(2H) + 2i == 2*(b*H + i) == 2*blockIdx.y, so batch/row never
// need to be separated.

typedef __attribute__((ext_vector_type(4))) float f32x4;

namespace {
constexpr unsigned Bn = 4;
constexpr unsigned H  = 384;
constexpr unsigned W  = 384;
constexpr unsigned C  = 64;
constexpr unsigned C4 = C / 4;            // 16 float4 chunks per pixel
constexpr unsigned OW = W * 2;            // 768
constexpr unsigned ROW_CHUNKS = W * C4;   // 6144 float4 chunks per input row
constexpr unsigned NROWS = Bn * H;        // 1536 input rows
}

__global__ __launch_bounds__(256)
void upsample2x2_nn_kernel(const f32x4* __restrict__ in, f32x4* __restrict__ out) {
    const unsigned row = blockIdx.y;                       // b*H + i   (0..1535)
    const unsigned t   = blockIdx.x * 256u + threadIdx.x;  // chunk in row (0..6143)
    const unsigned c4  = t & (C4 - 1);                     // channel chunk 0..15
    const unsigned j   = t >> 4;                           // input column

    // Each input float4 is read exactly once -> non-temporal load.
    const unsigned inIdx = row * ROW_CHUNKS + t;
    f32x4 v = __builtin_nontemporal_load(&in[inIdx]);

    // Output row pair starts at 2*row; horizontal positions 2j and 2j+1.
    const unsigned row0 = (2u * row) * (OW * C4) + ((j << 1) * C4 + c4);
    const unsigned row1 = row0 + OW * C4;

    // Output is write-once and >> 192 MB L2 -> non-temporal stores.
    __builtin_nontemporal_store(v, &out[row0]);        // (2i,   2j)
    __builtin_nontemporal_store(v, &out[row0 + C4]);   // (2i,   2j+1)
    __builtin_nontemporal_store(v, &out[row1]);        // (2i+1, 2j)
    __builtin_nontemporal_store(v, &out[row1 + C4]);   // (2i+1, 2j+1)
}

extern "C" void kernel_launch(void* const* d_in, const int* in_sizes, int n_in,
                              void* d_out, int out_size, void* d_ws, size_t ws_size,
                              hipStream_t stream) {
    (void)in_sizes; (void)n_in; (void)out_size; (void)d_ws; (void)ws_size;
    const f32x4* x = (const f32x4*)d_in[0];
    f32x4* out = (f32x4*)d_out;

    dim3 grid(ROW_CHUNKS / 256u, NROWS);   // (24, 1536) blocks, 256 thr each
    upsample2x2_nn_kernel<<<grid, dim3(256), 0, stream>>>(x, out);
}